// Mixup_31490700214323
// MI455X (gfx1250) — compile-verified
//
#include <hip/hip_runtime.h>
#include <hip/hip_bf16.h>

#define Nn 100000
#define Ee 1600000
#define Dd 128
#define Hh 128
#define Cc 40
#define Ll 3

typedef float v2f __attribute__((ext_vector_type(2)));
typedef float v8f __attribute__((ext_vector_type(8)));

// ---------------- utility kernels ----------------

__global__ __launch_bounds__(256) void k_zero(float* __restrict__ p, long long n) {
  long long i = (long long)blockIdx.x * 256 + threadIdx.x;
  long long stride = (long long)gridDim.x * 256;
  for (; i < n; i += stride) p[i] = 0.0f;
}

__global__ __launch_bounds__(256) void k_deg(const int* __restrict__ dst,
                                             float* __restrict__ deg) {
  int e = blockIdx.x * 256 + threadIdx.x;
  if (e < Ee) atomicAdd(&deg[dst[e]], 1.0f);
}

// Edge-parallel gather + scatter-add: one thread per (edge, 4-float chunk).
// Features are L2-resident (51MB << 192MB L2) so random access is cheap.
__global__ __launch_bounds__(256) void k_agg(const float* __restrict__ feat,
                                             const int* __restrict__ src,
                                             const int* __restrict__ dst,
                                             float* __restrict__ agg) {
  long long idx = (long long)blockIdx.x * 256 + threadIdx.x;
  if (idx >= (long long)Ee * 32) return;
  int e = (int)(idx >> 5);
  int q = (int)(idx & 31);
  int s = src[e];
  int d = dst[e];
  float4 v = *(const float4*)&feat[(size_t)s * Hh + q * 4];
  float* p = &agg[(size_t)d * Hh + q * 4];
  atomicAdd(p + 0, v.x);
  atomicAdd(p + 1, v.y);
  atomicAdd(p + 2, v.z);
  atomicAdd(p + 3, v.w);
}

__global__ __launch_bounds__(256) void k_perm(const float* __restrict__ in,
                                              const int* __restrict__ perm,
                                              float* __restrict__ out) {
  long long i = (long long)blockIdx.x * 256 + threadIdx.x;
  if (i >= (long long)Nn * Hh) return;
  int row = (int)(i / Hh);
  int col = (int)(i % Hh);
  out[i] = in[(size_t)perm[row] * Hh + col];
}

__global__ __launch_bounds__(256) void k_mix(const float* __restrict__ a,
                                             const float* __restrict__ b,
                                             const float* __restrict__ rp,
                                             float* __restrict__ out) {
  long long i = (long long)blockIdx.x * 256 + threadIdx.x;
  if (i >= (long long)Nn * Hh) return;
  float r = *rp;
  out[i] = a[i] * r + b[i] * (1.0f - r);
}

// ---------------- fused SAGE layer: out = relu(agg/deg @ Wl + bl + xdst @ Wr) ----
// Block = 256 threads = 8 waves. Block owns 16 nodes x full 128 hidden cols.
// A tiles (normalized agg + xdst, 16x128 each) staged in LDS; each wave does a
// 16-col tile via V_WMMA_F32_16X16X4_F32 (exact fp32, matches reference math).

__global__ __launch_bounds__(256) void k_sage(const float* __restrict__ agg,
                                              const float* __restrict__ deg,
                                              const float* __restrict__ xdst,
                                              const float* __restrict__ Wl,
                                              const float* __restrict__ bl,
                                              const float* __restrict__ Wr,
                                              float* __restrict__ out) {
  __shared__ float sA[16][Hh + 4];  // normalized aggregate (mean)
  __shared__ float sX[16][Hh + 4];  // destination features
  const int node0 = blockIdx.x * 16;
  const int tid = threadIdx.x;

  // Cooperative fill: 16 rows x 32 float4 per array, 2 iterations of 256 threads.
  for (int i = tid; i < 16 * 32; i += 256) {
    int m = i >> 5;
    int q = i & 31;
    int node = node0 + m;
    float4 va = {0.f, 0.f, 0.f, 0.f};
    float4 vx = {0.f, 0.f, 0.f, 0.f};
    if (node < Nn) {
      va = *(const float4*)&agg[(size_t)node * Hh + q * 4];
      vx = *(const float4*)&xdst[(size_t)node * Hh + q * 4];
      float inv = 1.0f / fmaxf(deg[node], 1.0f);  // mean aggregation
      va.x *= inv; va.y *= inv; va.z *= inv; va.w *= inv;
    }
    *(float4*)&sA[m][q * 4] = va;
    *(float4*)&sX[m][q * 4] = vx;
  }
  __syncthreads();

  const int wave = tid >> 5;          // 0..7 -> which 16-col tile of H
  const int lane = tid & 31;
  const int m    = lane & 15;         // A-fragment row (per ISA layout)
  const int krow = (lane >> 4) * 2;   // lanes 0-15: K={0,1}; lanes 16-31: K={2,3}
  const int col  = wave * 16 + (lane & 15);

  v8f c;
  float bias = bl[col];
  #pragma unroll
  for (int v = 0; v < 8; ++v) c[v] = bias;

  const float* Bl = Wl;
  const float* Br = Wr;

  // agg_norm @ Wl
  #pragma unroll 4
  for (int k0 = 0; k0 < Hh; k0 += 4) {
    v2f a, b;
    float2 av = *(const float2*)&sA[m][k0 + krow];
    a.x = av.x; a.y = av.y;
    b.x = Bl[(size_t)(k0 + krow) * Hh + col];
    b.y = Bl[(size_t)(k0 + krow + 1) * Hh + col];
    c = __builtin_amdgcn_wmma_f32_16x16x4_f32(false, a, false, b, (short)0, c,
                                              false, false);
  }
  // xdst @ Wr
  #pragma unroll 4
  for (int k0 = 0; k0 < Hh; k0 += 4) {
    v2f a, b;
    float2 av = *(const float2*)&sX[m][k0 + krow];
    a.x = av.x; a.y = av.y;
    b.x = Br[(size_t)(k0 + krow) * Hh + col];
    b.y = Br[(size_t)(k0 + krow + 1) * Hh + col];
    c = __builtin_amdgcn_wmma_f32_16x16x4_f32(false, a, false, b, (short)0, c,
                                              false, false);
  }

  // C/D layout: VGPR v, lanes 0-15 -> M=v, lanes 16-31 -> M=v+8; N = lane%16
  const int mrow = (lane >> 4) * 8;
  #pragma unroll
  for (int v = 0; v < 8; ++v) {
    int node = node0 + v + mrow;
    if (node < Nn) out[(size_t)node * Hh + col] = fmaxf(c[v], 0.0f);
  }
}

// ---------------- output head: logits + log_softmax ----------------

__global__ __launch_bounds__(64) void k_out(const float* __restrict__ xmix,
                                            const float* __restrict__ Wo,
                                            const float* __restrict__ bo,
                                            float* __restrict__ out) {
  __shared__ float vals[Cc];
  __shared__ float red[2];
  const int n = blockIdx.x;
  const int c = threadIdx.x;
  if (c < Cc) {
    float acc = bo[c];
    const float* xr = &xmix[(size_t)n * Hh];
    #pragma unroll 4
    for (int k = 0; k < Hh; ++k) acc += xr[k] * Wo[k * Cc + c];
    vals[c] = acc;
  }
  __syncthreads();
  if (c == 0) {
    float mx = vals[0];
    for (int i = 1; i < Cc; ++i) mx = fmaxf(mx, vals[i]);
    float s = 0.0f;
    for (int i = 0; i < Cc; ++i) s += expf(vals[i] - mx);
    red[0] = mx;
    red[1] = logf(s);
  }
  __syncthreads();
  if (c < Cc) out[(size_t)n * Cc + c] = vals[c] - red[0] - red[1];
}

// ---------------- driver ----------------

static inline void run_sage(const float* src_feat, const float* xdst,
                            const int* src, const int* dst, const float* deg,
                            const float* Wl, const float* bl, const float* Wr,
                            float* agg, float* out, hipStream_t stream) {
  const long long NH = (long long)Nn * Hh;
  k_zero<<<2048, 256, 0, stream>>>(agg, NH);
  const long long work = (long long)Ee * 32;
  k_agg<<<(unsigned)((work + 255) / 256), 256, 0, stream>>>(src_feat, src, dst, agg);
  k_sage<<<(Nn + 15) / 16, 256, 0, stream>>>(agg, deg, xdst, Wl, bl, Wr, out);
}

extern "C" void kernel_launch(void* const* d_in, const int* in_sizes, int n_in,
                              void* d_out, int out_size, void* d_ws, size_t ws_size,
                              hipStream_t stream) {
  const float* x     = (const float*)d_in[0];
  const float* Wl    = (const float*)d_in[1];  // [L,128,128]
  const float* bl    = (const float*)d_in[2];  // [L,128]
  const float* Wr    = (const float*)d_in[3];  // [L,128,128]
  const float* Wo    = (const float*)d_in[4];  // [128,40]
  const float* bo    = (const float*)d_in[5];  // [40]
  const float* r     = (const float*)d_in[6];  // scalar
  const int*   adj   = (const int*)d_in[7];    // [2,E]
  const int*   adjb  = (const int*)d_in[8];    // [2,E]
  const int*   perm  = (const int*)d_in[9];    // [N]

  const int *srcA = adj,  *dstA = adj + Ee;
  const int *srcB = adjb, *dstB = adjb + Ee;

  const size_t NH = (size_t)Nn * Hh;
  float* ws   = (float*)d_ws;
  float* deg  = ws;            // N
  float* degb = ws + Nn;       // N
  float* agg  = ws + 2 * Nn;   // N*H scratch
  float* a1   = agg  + NH;
  float* a2   = a1   + NH;
  float* b0   = a2   + NH;
  float* b1   = b0   + NH;
  float* b2   = b1   + NH;
  float* xmix = b2   + NH;
  float* xnew = xmix + NH;
  float* xnwb = xnew + NH;

  // degrees (constant across layers, computed once per launch)
  k_zero<<<(Nn + 255) / 256, 256, 0, stream>>>(deg, Nn);
  k_zero<<<(Nn + 255) / 256, 256, 0, stream>>>(degb, Nn);
  k_deg<<<(Ee + 255) / 256, 256, 0, stream>>>(dstA, deg);
  k_deg<<<(Ee + 255) / 256, 256, 0, stream>>>(dstB, degb);

  // pre-aggregation passes: a1 = relu(sage(x,x)), a2 = relu(sage(a1,a1))
  run_sage(x,  x,  srcA, dstA, deg, Wl + 0 * Dd * Hh, bl + 0 * Hh,
           Wr + 0 * Dd * Hh, agg, a1, stream);
  run_sage(a1, a1, srcA, dstA, deg, Wl + 1 * Dd * Hh, bl + 1 * Hh,
           Wr + 1 * Dd * Hh, agg, a2, stream);

  // permuted branch features
  const unsigned pgrid = (unsigned)((NH + 255) / 256);
  k_perm<<<pgrid, 256, 0, stream>>>(x,  perm, b0);
  k_perm<<<pgrid, 256, 0, stream>>>(a1, perm, b1);
  k_perm<<<pgrid, 256, 0, stream>>>(a2, perm, b2);

  // x_mix = x*r + b0*(1-r)
  k_mix<<<pgrid, 256, 0, stream>>>(x, b0, r, xmix);

  const float* srcF[3] = {x, a1, a2};
  const float* srcFb[3] = {b0, b1, b2};
  for (int i = 0; i < Ll; ++i) {
    const float* Wli = Wl + (size_t)i * Dd * Hh;
    const float* bli = bl + (size_t)i * Hh;
    const float* Wri = Wr + (size_t)i * Dd * Hh;
    run_sage(srcF[i],  xmix, srcA, dstA, deg,  Wli, bli, Wri, agg, xnew, stream);
    run_sage(srcFb[i], xmix, srcB, dstB, degb, Wli, bli, Wri, agg, xnwb, stream);
    k_mix<<<pgrid, 256, 0, stream>>>(xnew, xnwb, r, xmix);
  }

  // logits + log_softmax
  k_out<<<Nn, 64, 0, stream>>>(xmix, Wo, bo, (float*)d_out);
}